// VariationalEncoder_13288628814616
// MI455X (gfx1250) — compile-verified
//
#include <hip/hip_runtime.h>
#include <hip/hip_bf16.h>

#define N_NODES 50000
#define IN_C    256
#define HID     256
#define OUT_C   128

typedef __attribute__((ext_vector_type(2))) float v2f;
typedef __attribute__((ext_vector_type(8))) float v8f;

// ---------------- degree / normalization ----------------

__global__ void deg_init_kernel(float* __restrict__ deg, int n) {
  int i = blockIdx.x * blockDim.x + threadIdx.x;
  if (i < n) deg[i] = 1.0f;  // self-loop contributes 1 to every node's degree
}

__global__ void deg_accum_kernel(const long long* __restrict__ dst,
                                 float* __restrict__ deg, int E) {
  int e = blockIdx.x * blockDim.x + threadIdx.x;
  if (e < E) atomicAdd(&deg[(int)dst[e]], 1.0f);
}

__global__ void deg_rsqrt_kernel(float* __restrict__ deg, int n) {
  int i = blockIdx.x * blockDim.x + threadIdx.x;
  if (i < n) {
    float d = deg[i];
    deg[i] = (d > 0.0f) ? rsqrtf(d) : 0.0f;  // becomes dinv in-place
  }
}

// ---------------- fp32 WMMA GEMM with LDS-staged A tile ----------------
// C[M,N] = A[M,K] * B[K,N], row-major. K, N compile-time so the K-loop fully
// unrolls (immediate-offset loads, no per-iteration address math).
// Block = 128 threads (4 waves), one block per 16-row M tile:
//   - cooperative float4 load of the 16xK A tile into LDS (row stride K+4
//     floats -> ds_load_b64 fragment reads are bank-conflict free, and
//     (K+4)*4 = 1040 B rows keep 16B alignment for b128 stores)
//   - each wave loops over N/16 tiles (stride 4), reusing the LDS A tile.
// Fragments per the CDNA5 ISA 16x16x4 f32 layout:
//   A (16x4, 2 VGPR): lanes 0-15 -> K=k,k+1 ; lanes 16-31 -> K=k+2,k+3
//   B (4x16, 2 VGPR): mirror with N = lane&15
//   C (16x16, 8 VGPR): lanes 0-15 -> M=0..7, lanes 16-31 -> M=8..15
template <int K, int N>
__global__ void __launch_bounds__(128)
wmma_gemm_f32_lds_kernel(const float* __restrict__ A,
                         const float* __restrict__ B,
                         float* __restrict__ C) {
  constexpr int LDSW = K + 4;            // padded row stride (floats)
  __shared__ float As[16 * LDSW];

  const int t     = threadIdx.x;
  const int lane  = t & 31;
  const int wave  = t >> 5;
  const int tileM = blockIdx.x;

  // ---- cooperative load: 16 rows x K floats = 16*K/4 float4s over 128 threads
  {
    const float4* A4 = (const float4*)(A + (size_t)tileM * 16 * K);
    constexpr int NV4 = 16 * K / 4;
#pragma unroll
    for (int i = 0; i < NV4 / 128; ++i) {
      int flat = t + i * 128;
      int row  = flat / (K / 4);
      int c4   = flat % (K / 4);
      float4 v = A4[row * (K / 4) + c4];
      *(float4*)&As[row * LDSW + c4 * 4] = v;
    }
  }
  __syncthreads();

  const int mn    = lane & 15;           // M for A frag, N for B frag
  const int khalf = (lane >> 4) << 1;    // 0 (lanes 0-15) or 2 (lanes 16-31)
  const int mbase = (lane >> 4) << 3;    // 0 or 8 for C store
  const float* Alds = &As[mn * LDSW + khalf];

  for (int tn = wave; tn < N / 16; tn += 4) {
    const float* Bbase = B + tn * 16 + mn;
    v8f c = {};
#pragma unroll
    for (int k = 0; k < K; k += 4) {
      v2f a, b;
      a[0] = Alds[k];                     // merges into ds_load_b64
      a[1] = Alds[k + 1];
      b[0] = Bbase[(k + khalf) * N];      // immediate-offset global loads
      b[1] = Bbase[(k + khalf + 1) * N];
      // 8 args: (neg_a, A, neg_b, B, c_mod, C, reuse_a, reuse_b)
      c = __builtin_amdgcn_wmma_f32_16x16x4_f32(false, a, false, b,
                                                (short)0, c, false, false);
    }
    float* Cp = C + ((size_t)tileM * 16 + mbase) * N + tn * 16 + mn;
#pragma unroll
    for (int i = 0; i < 8; ++i) Cp[i * N] = c[i];
  }
}

// ---------------- aggregation (self-loop init, edge scatter, bias/act) ----------------

// out[i,:] = h[i,:] * dinv[i]^2   (the self-loop message)
__global__ void self_msg_kernel(const float* __restrict__ h,
                                const float* __restrict__ dinv,
                                float* __restrict__ out, int F) {
  int i = blockIdx.x;
  int f = threadIdx.x;
  float nrm = dinv[i] * dinv[i];
  size_t idx = (size_t)i * F + f;
  out[idx] = h[idx] * nrm;
}

// one block per edge; coalesced gather of h[src,:], coalesced atomic adds into out[dst,:]
__global__ void scatter_kernel(const float* __restrict__ h,
                               const float* __restrict__ dinv,
                               const long long* __restrict__ src,
                               const long long* __restrict__ dst,
                               float* __restrict__ out, int F) {
  int e = blockIdx.x;
  int f = threadIdx.x;
  int s = (int)src[e];
  int d = (int)dst[e];
  float nrm = dinv[s] * dinv[d];
  atomicAdd(&out[(size_t)d * F + f], h[(size_t)s * F + f] * nrm);
}

__global__ void bias_act_kernel(const float* __restrict__ in,
                                const float* __restrict__ bias,
                                float* __restrict__ out, int F, int do_relu) {
  int i = blockIdx.x;
  int f = threadIdx.x;
  size_t idx = (size_t)i * F + f;
  float v = in[idx] + bias[f];
  out[idx] = do_relu ? fmaxf(v, 0.0f) : v;
}

// ---------------- driver ----------------

extern "C" void kernel_launch(void* const* d_in, const int* in_sizes, int n_in,
                              void* d_out, int out_size, void* d_ws, size_t ws_size,
                              hipStream_t stream) {
  const float*     x    = (const float*)d_in[0];
  const long long* ei   = (const long long*)d_in[1];  // int64 [2, E]
  const float*     W1   = (const float*)d_in[2];      // [256, 256]
  const float*     b1   = (const float*)d_in[3];
  const float*     Wmu  = (const float*)d_in[4];      // [256, 128]
  const float*     bmu  = (const float*)d_in[5];
  const float*     Wlv  = (const float*)d_in[6];      // [256, 128]
  const float*     blv  = (const float*)d_in[7];

  const int n = N_NODES;
  const int E = in_sizes[1] / 2;
  const long long* esrc = ei;       // edge_index[0]
  const long long* edst = ei + E;   // edge_index[1]

  // workspace layout (floats): dinv[n] | h1[n*HID] | agg[n*HID] (reused as tmu/tlv)
  float* ws   = (float*)d_ws;
  float* dinv = ws;
  float* h1   = dinv + n;
  float* agg  = h1 + (size_t)n * HID;
  float* tmu  = agg;                          // reuse after layer-1 aggregation done
  float* tlv  = agg + (size_t)n * OUT_C;
  float* out_mu = (float*)d_out;
  float* out_lv = out_mu + (size_t)n * OUT_C;

  // --- symmetric normalization factors (deg = in-degree + 1, dinv = rsqrt) ---
  deg_init_kernel <<<(n + 255) / 256, 256, 0, stream>>>(dinv, n);
  deg_accum_kernel<<<(E + 255) / 256, 256, 0, stream>>>(edst, dinv, E);
  deg_rsqrt_kernel<<<(n + 255) / 256, 256, 0, stream>>>(dinv, n);

  // --- layer 1: h1 = x @ W1 ; agg = scatter ; h1 = relu(agg + b1) ---
  wmma_gemm_f32_lds_kernel<IN_C, HID><<<n / 16, 128, 0, stream>>>(x, W1, h1);
  self_msg_kernel<<<n, HID, 0, stream>>>(h1, dinv, agg, HID);
  scatter_kernel <<<E, HID, 0, stream>>>(h1, dinv, esrc, edst, agg, HID);
  bias_act_kernel<<<n, HID, 0, stream>>>(agg, b1, h1, HID, 1);  // h1 = relu(...)

  // --- mu head: tmu = h1 @ Wmu ; aggregate directly into d_out ; + bmu ---
  wmma_gemm_f32_lds_kernel<HID, OUT_C><<<n / 16, 128, 0, stream>>>(h1, Wmu, tmu);
  self_msg_kernel<<<n, OUT_C, 0, stream>>>(tmu, dinv, out_mu, OUT_C);
  scatter_kernel <<<E, OUT_C, 0, stream>>>(tmu, dinv, esrc, edst, out_mu, OUT_C);
  bias_act_kernel<<<n, OUT_C, 0, stream>>>(out_mu, bmu, out_mu, OUT_C, 0);

  // --- logvar head: tlv = h1 @ Wlv ; aggregate into d_out ; + blv ---
  wmma_gemm_f32_lds_kernel<HID, OUT_C><<<n / 16, 128, 0, stream>>>(h1, Wlv, tlv);
  self_msg_kernel<<<n, OUT_C, 0, stream>>>(tlv, dinv, out_lv, OUT_C);
  scatter_kernel <<<E, OUT_C, 0, stream>>>(tlv, dinv, esrc, edst, out_lv, OUT_C);
  bias_act_kernel<<<n, OUT_C, 0, stream>>>(out_lv, blv, out_lv, OUT_C, 0);
}